// RPTHNet_78262894067737
// MI455X (gfx1250) — compile-verified
//
#include <hip/hip_runtime.h>

// RPTH Newton solver: B=4096 independent SPD systems, n=128, 10 iterations.
// One workgroup (256 threads = 8 wave32) per batch.
//  - Q[b] staged into LDS ONCE via the Tensor Data Mover (tensor_load_to_lds),
//    using D# pad_enable to land directly in the padded stride-132 layout.
//  - Per-iteration h rebuild is an LDS->LDS copy; Cholesky trailing updates
//    run on V_WMMA_F32_16X16X4_F32.

#define NDIM 128
#define LSTR 132            // padded LDS row stride (floats); 132*4=528B, 16B-aligned rows
#define TB   16
#define NT   (NDIM / TB)    // 8 tile rows/cols
#define MAX_ITERS 10
#define FLCONST 0.3627016654f   // 0.95*(3-sqrt(5))/2

typedef __attribute__((ext_vector_type(2))) float v2f;
typedef __attribute__((ext_vector_type(8))) float v8f;
typedef __attribute__((ext_vector_type(4))) unsigned int u32x4;
typedef __attribute__((ext_vector_type(8))) int i32x8;
typedef __attribute__((ext_vector_type(4))) int i32x4;

#if defined(__has_builtin)
#if __has_builtin(__builtin_amdgcn_tensor_load_to_lds)
#define HAVE_TDM 1
#endif
#endif

__launch_bounds__(256)
__global__ void rpth_solve_kernel(const float* __restrict__ Q,
                                  const float* __restrict__ r,
                                  float* __restrict__ out)
{
    __shared__ float sQ[NDIM * LSTR];   // Q[b], LDS-resident for all iterations
    __shared__ float sH[NDIM * LSTR];   // h / Cholesky factor (lower)
    __shared__ float sx[NDIM];
    __shared__ float sr[NDIM];          // |r|
    __shared__ float su[NDIM];          // u = Qx - r/x
    __shared__ float sd[NDIM];          // solve workspace -> dk
    __shared__ float sScal[4];          // 0: gk, 1: lk_prev, 3: xsum

    const int tid  = threadIdx.x;
    const int lane = tid & 31;
    const int wave = tid >> 5;
    const int b    = blockIdx.x;

    const float* Qb = Q + (size_t)b * NDIM * NDIM;
    const float* rb = r + (size_t)b * NDIM;

    // ---- stage Q[b] -> sQ (padded stride LSTR) via TDM ----
#ifdef HAVE_TDM
    if (tid == 0) {
        unsigned long long ga = (unsigned long long)(size_t)Qb;
        unsigned int ldsoff   = (unsigned int)(size_t)(void*)&sQ[0];
        u32x4 g0;
        g0.x = 1u;                                  // count=1 (valid user D#)
        g0.y = ldsoff;                              // lds_addr (bytes)
        g0.z = (unsigned int)ga;                    // global_addr[31:0]
        g0.w = ((unsigned int)(ga >> 32) & 0x01FFFFFFu) | (2u << 30); // ga[56:32] | type=2
        i32x8 g1;
        // data_size=2 (4B) | pad_enable | pad_interval=6 (128 DW) | pad_amount=3 (4 DW)
        g1.s0 = (int)((2u << 16) | (1u << 20) | (6u << 22) | (3u << 25));
        g1.s1 = (int)(128u << 16);                  // tensor_dim0 = 128 (low16 @ [31:16])
        g1.s2 = (int)(128u << 16);                  // tensor_dim0 hi=0 | tensor_dim1=128 lo
        g1.s3 = (int)(128u << 16);                  // tensor_dim1 hi=0 | tile_dim0=128
        g1.s4 = 128;                                // tile_dim1=128 | tile_dim2=0
        g1.s5 = 128;                                // tensor_dim0_stride = 128 (low32)
        g1.s6 = 0;                                  // stride hi | tensor_dim1_stride lo
        g1.s7 = 0;
        i32x4 g2 = {0, 0, 0, 0};
        i32x4 g3 = {0, 0, 0, 0};
#if defined(__clang_major__) && (__clang_major__ >= 23)
        i32x8 g4 = {0, 0, 0, 0, 0, 0, 0, 0};
        __builtin_amdgcn_tensor_load_to_lds(g0, g1, g2, g3, g4, 0);
#else
        __builtin_amdgcn_tensor_load_to_lds(g0, g1, g2, g3, 0);
#endif
        __builtin_amdgcn_s_wait_tensorcnt(0);
    }
#else
    {
        const float4* Q4 = (const float4*)Qb;
        for (int i = tid; i < NDIM * NDIM / 4; i += 256) {
            float4 v = Q4[i];
            int e = i * 4;
            int rr = e >> 7, cc = e & (NDIM - 1);
            *(float4*)(&sQ[rr * LSTR + cc]) = v;
        }
    }
#endif
    __syncthreads();

    // ---- init: x0 = |r| / sqrt(diag(Q)) ----
    if (tid < NDIM) {
        float rv = fabsf(rb[tid]);
        float qd = sQ[tid * LSTR + tid];
        sr[tid]  = rv;
        sx[tid]  = rv * rsqrtf(qd);
    }
    if (tid == 0) sScal[1] = FLCONST + 1.0f;   // lk_prev
    __syncthreads();

    for (int it = 0; it < MAX_ITERS; ++it) {
        // ---- u = Q x - r/x : one row per thread (tid < 128), from LDS ----
        if (tid < NDIM) {
            const float4* row4 = (const float4*)(&sQ[tid * LSTR]);
            float acc = 0.f;
            #pragma unroll 8
            for (int j4 = 0; j4 < NDIM / 4; ++j4) {
                float4 q = row4[j4];
                int j = j4 * 4;
                acc += q.x * sx[j] + q.y * sx[j + 1] + q.z * sx[j + 2] + q.w * sx[j + 3];
            }
            su[tid] = acc - sr[tid] / sx[tid];
        }
        // ---- h = Q (LDS->LDS copy), then fix diagonal ----
        for (int i = tid; i < NDIM * NDIM / 4; i += 256) {
            int e = i * 4;
            int rr = e >> 7, cc = e & (NDIM - 1);
            *(float4*)(&sH[rr * LSTR + cc]) = *(const float4*)(&sQ[rr * LSTR + cc]);
        }
        __syncthreads();
        if (tid < NDIM) {
            float xi = sx[tid];
            sH[tid * LSTR + tid] = sQ[tid * LSTR + tid] + sr[tid] / (xi * xi);
        }
        __syncthreads();

        // ---- blocked Cholesky (lower), panel width 16 ----
        for (int p = 0; p < NT; ++p) {
            const int pk = p * TB;

            // panel factor (unblocked right-looking over 16 columns)
            for (int k = 0; k < TB; ++k) {
                const int c = pk + k;
                __syncthreads();                       // prior rank-1 complete
                float hcc = sH[c * LSTR + c];
                __syncthreads();                       // all read before pivot overwrite
                float piv = sqrtf(hcc);
                float inv = 1.0f / piv;
                if (tid == 0) sH[c * LSTR + c] = piv;
                int i = c + 1 + tid;
                float lic = 0.f;
                if (i < NDIM) {
                    lic = sH[i * LSTR + c] * inv;
                    sH[i * LSTR + c] = lic;
                }
                __syncthreads();                       // scaled column visible
                if (i < NDIM) {
                    const int jend = pk + TB;
                    for (int j = c + 1; j < jend; ++j)
                        if (j <= i) sH[i * LSTR + j] -= lic * sH[j * LSTR + c];
                }
            }
            __syncthreads();

            // trailing update: C(ti,tj) -= L(ti,p) * L(tj,p)^T  via f32 WMMA
            const int nti    = NT - 1 - p;
            const int nTiles = nti * (nti + 1) / 2;
            for (int t = wave; t < nTiles; t += 8) {
                int q = 0, rem = t;                    // row-block q has q+1 tiles
                while (rem > q) { rem -= (q + 1); ++q; }
                const int ti = p + 1 + q;
                const int tj = p + 1 + rem;

                const int m  = lane & 15;              // M / N index within tile
                const int kh = lane >> 4;              // lane-half selects K pair
                const int crb = ti * TB + (kh ? 8 : 0);
                const int ccc = tj * TB + m;

                v8f c;
                #pragma unroll
                for (int v = 0; v < 8; ++v) c[v] = sH[(crb + v) * LSTR + ccc];

                const int arow = (ti * TB + m) * LSTR;
                const int brow = (tj * TB + m) * LSTR;
                #pragma unroll
                for (int kc = 0; kc < 4; ++kc) {
                    const int k0 = pk + kc * 4 + 2 * kh;
                    v2f av, bv;
                    av.x = -sH[arow + k0];             // negate A: D = (-A)B + C
                    av.y = -sH[arow + k0 + 1];
                    bv.x =  sH[brow + k0];             // B = L(tj)^T, lane = column
                    bv.y =  sH[brow + k0 + 1];
                    c = __builtin_amdgcn_wmma_f32_16x16x4_f32(
                            false, av, false, bv, (short)0, c, false, false);
                }
                #pragma unroll
                for (int v = 0; v < 8; ++v) sH[(crb + v) * LSTR + ccc] = c[v];
            }
            __syncthreads();
        }

        // ---- forward solve: L y = u   (y in sd) ----
        if (tid < NDIM) sd[tid] = su[tid];
        __syncthreads();
        for (int c = 0; c < NDIM; ++c) {
            if (tid == 0) sd[c] = sd[c] / sH[c * LSTR + c];
            __syncthreads();
            int i = c + 1 + tid;
            if (i < NDIM) sd[i] -= sH[i * LSTR + c] * sd[c];
            __syncthreads();
        }
        // ---- backward solve: L^T d = y ----
        for (int c = NDIM - 1; c >= 0; --c) {
            if (tid == 0) sd[c] = sd[c] / sH[c * LSTR + c];
            __syncthreads();
            if (tid < c) sd[tid] -= sH[c * LSTR + tid] * sd[c];
            __syncthreads();
        }

        // ---- gk, lk, x update ----
        if (tid == 0) {
            float g = 0.f, l2 = 0.f;
            for (int i = 0; i < NDIM; ++i) {
                float dv = sd[i];
                g  = fmaxf(g, fabsf(dv / sx[i]));
                l2 += dv * su[i];
            }
            float lk_prev = sScal[1];
            sScal[0] = (lk_prev <= FLCONST) ? 0.f : g;  // gk
            sScal[1] = sqrtf(l2);                       // lk_new
        }
        __syncthreads();
        if (tid < NDIM) {
            float gk = sScal[0];
            sx[tid] = sx[tid] - sd[tid] / (1.0f + gk);
        }
        __syncthreads();
    }

    // ---- normalize: out = x / (sum|x| + 1e-8) ----
    if (tid == 0) {
        float s = 0.f;
        for (int i = 0; i < NDIM; ++i) s += fabsf(sx[i]);
        sScal[3] = s + 1e-8f;
    }
    __syncthreads();
    if (tid < NDIM)
        out[(size_t)b * NDIM + tid] = sx[tid] / sScal[3];
}

extern "C" void kernel_launch(void* const* d_in, const int* in_sizes, int n_in,
                              void* d_out, int out_size, void* d_ws, size_t ws_size,
                              hipStream_t stream) {
    const float* Q = (const float*)d_in[0];   // [B,128,128] f32
    const float* r = (const float*)d_in[1];   // [B,128,1]   f32
    float* out = (float*)d_out;               // [B,128,1]   f32
    const int Bc = in_sizes[1] / NDIM;        // batch count from r
    rpth_solve_kernel<<<dim3(Bc), dim3(256), 0, stream>>>(Q, r, out);
}